// EventSequenceEmbedder_14843406975105
// MI455X (gfx1250) — compile-verified
//
#include <hip/hip_runtime.h>
#include <hip/hip_bf16.h>

#define DM 256      // d_model
#define BB 32       // batch
#define EE 256      // events
#define CC 7        // cards per event
#define MP 9
#define NA 8

typedef __attribute__((ext_vector_type(2))) float v2f;
typedef __attribute__((ext_vector_type(8))) float v8f;

// ---------------------------------------------------------------------------
// Kernel 1: fused-table GEMM  C[M,256] = A[M,256] @ B[256,256]
// One wave (32 threads) per 16x16 output tile, v_wmma_f32_16x16x4_f32 over K.
// f32 WMMA VGPR layout (ISA 7.12.2):
//   A 16x4:  lane L holds row M=L%16; VGPR v holds K = k0 + v + 2*(L>=16)
//   B 4x16:  lane L holds col N=L%16; VGPR v holds K = k0 + v + 2*(L>=16)
//   C/D:     VGPR r holds row M = r + 8*(L>=16), col N = L%16
// Rows >= M are zero-padded (clamped address * 0 scale; no EXEC divergence).
// ---------------------------------------------------------------------------
__global__ __launch_bounds__(32) void fused_table_gemm(
    const float* __restrict__ A, const float* __restrict__ B,
    float* __restrict__ C, int M)
{
    const int K = DM, N = DM;
    const int tm = blockIdx.x, tn = blockIdx.y;
    const int lane = threadIdx.x & 31;
    const int half = lane >> 4;
    const int l16  = lane & 15;

    const int am   = tm * 16 + l16;
    const int amc  = (am < M) ? am : (M - 1);
    const float as = (am < M) ? 1.0f : 0.0f;
    const int bn   = tn * 16 + l16;

    const float* __restrict__ arow = A + (size_t)amc * K;

    v8f acc = {0.f, 0.f, 0.f, 0.f, 0.f, 0.f, 0.f, 0.f};
    for (int k0 = 0; k0 < K; k0 += 4) {
        const int ka = k0 + 2 * half;
        v2f a, b;
        a.x = arow[ka + 0] * as;
        a.y = arow[ka + 1] * as;
        b.x = B[(size_t)(ka + 0) * N + bn];
        b.y = B[(size_t)(ka + 1) * N + bn];
        acc = __builtin_amdgcn_wmma_f32_16x16x4_f32(
            /*neg_a=*/false, a, /*neg_b=*/false, b,
            /*c_mod=*/(short)0, acc, /*reuse_a=*/false, /*reuse_b=*/false);
    }

    #pragma unroll
    for (int r = 0; r < 8; ++r) {
        const int m = tm * 16 + r + 8 * half;
        if (m < M) C[(size_t)m * N + tn * 16 + l16] = acc[r];
    }
}

// ---------------------------------------------------------------------------
// Kernel 2: fused bias  b_total = b_combine + b_scalar@W4 + b_bet@W5 + b_action@W6
// Tiny (256 threads x 768 MACs); VALU is fine.
// ---------------------------------------------------------------------------
__global__ __launch_bounds__(256) void fuse_bias(
    const float* __restrict__ b_combine, const float* __restrict__ b_scalar,
    const float* __restrict__ b_bet,     const float* __restrict__ b_action,
    const float* __restrict__ Wcomb,     float* __restrict__ b_total)
{
    const int o = threadIdx.x;
    const float* W4 = Wcomb + (size_t)1024 * DM;
    const float* W5 = Wcomb + (size_t)1280 * DM;
    const float* W6 = Wcomb + (size_t)1536 * DM;
    float v = b_combine[o];
    for (int i = 0; i < DM; ++i) v += b_scalar[i] * W4[(size_t)i * DM + o];
    for (int i = 0; i < DM; ++i) v += b_bet[i]    * W5[(size_t)i * DM + o];
    for (int i = 0; i < DM; ++i) v += b_action[i] * W6[(size_t)i * DM + o];
    b_total[o] = v;
}

// ---------------------------------------------------------------------------
// Kernel 3: main pass. One block = one event (b,e); 7 waves, wave c owns
// card c's 256-channel row (8 channels per lane, stride-32). LayerNorm
// reduces within a single wave32 via __shfl_xor.
// ---------------------------------------------------------------------------
__global__ __launch_bounds__(224) void event_embed(
    const int* __restrict__ card_ids, const int* __restrict__ hero_pos,
    const int* __restrict__ acting_pos, const int* __restrict__ num_players,
    const int* __restrict__ seq_lengths,
    const float* __restrict__ scalars, const float* __restrict__ bets,
    const float* __restrict__ action,
    const float* __restrict__ Tc, const float* __restrict__ Th,
    const float* __restrict__ Ta, const float* __restrict__ Tn,
    const float* __restrict__ Ws2, const float* __restrict__ Wb2,
    const float* __restrict__ Wa2, const float* __restrict__ b_total,
    const float* __restrict__ source_table,
    const float* __restrict__ ln_gamma, const float* __restrict__ ln_beta,
    float* __restrict__ out_emb, float* __restrict__ out_mask)
{
    const int ev   = blockIdx.x;          // 0 .. B*E-1
    const int b    = ev >> 8;
    const int e    = ev & 255;
    const int c    = threadIdx.x >> 5;    // wave index == card index
    const int lane = threadIdx.x & 31;

    const int hero = hero_pos[ev];
    const int act  = acting_pos[ev];
    const int np   = num_players[ev];
    const int card = card_ids[ev * CC + c];
    const float maskv = (e < seq_lengths[b]) ? 1.0f : 0.0f;

    const float s0 = scalars[ev * 2 + 0];
    const float s1 = scalars[ev * 2 + 1];
    float bet[MP];
    #pragma unroll
    for (int i = 0; i < MP; ++i) bet[i] = bets[ev * MP + i];
    float av[NA];
    #pragma unroll
    for (int i = 0; i < NA; ++i) av[i] = action[ev * NA + i];

    float h[8];
    float mean = 0.0f;
    #pragma unroll
    for (int j = 0; j < 8; ++j) {
        const int d = j * 32 + lane;
        float v = b_total[d]
                + Tc[(size_t)card * DM + d]
                + Th[(size_t)hero * DM + d]
                + Ta[(size_t)act  * DM + d]
                + Tn[(size_t)np   * DM + d];
        v = fmaf(s0, Ws2[0 * DM + d], v);
        v = fmaf(s1, Ws2[1 * DM + d], v);
        #pragma unroll
        for (int i = 0; i < MP; ++i) v = fmaf(bet[i], Wb2[i * DM + d], v);
        #pragma unroll
        for (int i = 0; i < NA; ++i) v = fmaf(av[i], Wa2[i * DM + d], v);
        h[j] = v;
        mean += v;
    }

    // wave32 reductions
    #pragma unroll
    for (int off = 16; off > 0; off >>= 1) mean += __shfl_xor(mean, off, 32);
    mean *= (1.0f / 256.0f);

    float var = 0.0f;
    #pragma unroll
    for (int j = 0; j < 8; ++j) { const float t = h[j] - mean; var += t * t; }
    #pragma unroll
    for (int off = 16; off > 0; off >>= 1) var += __shfl_xor(var, off, 32);
    var *= (1.0f / 256.0f);
    const float rstd = __frsqrt_rn(var + 1e-5f);

    const int src = (c >= 5) ? 1 : 0;     // source_ids = [0,0,0,0,0,1,1]
    const size_t row = (size_t)ev * CC + c;
    #pragma unroll
    for (int j = 0; j < 8; ++j) {
        const int d = j * 32 + lane;
        const float o = ((h[j] - mean) * rstd * ln_gamma[d] + ln_beta[d]
                         + source_table[(size_t)src * DM + d]) * maskv;
        out_emb[row * DM + d] = o;
    }
    if (lane == 0) out_mask[row] = maskv;
}

// ---------------------------------------------------------------------------
extern "C" void kernel_launch(void* const* d_in, const int* in_sizes, int n_in,
                              void* d_out, int out_size, void* d_ws, size_t ws_size,
                              hipStream_t stream) {
    const int*   card_ids     = (const int*)  d_in[0];
    const int*   hero_pos     = (const int*)  d_in[1];
    const int*   acting_pos   = (const int*)  d_in[2];
    const int*   num_players  = (const int*)  d_in[3];
    const int*   seq_lengths  = (const int*)  d_in[4];
    const float* scalars      = (const float*)d_in[5];
    const float* bets         = (const float*)d_in[6];
    const float* action       = (const float*)d_in[7];
    const float* card_table   = (const float*)d_in[8];
    const float* source_table = (const float*)d_in[9];
    const float* hero_table   = (const float*)d_in[10];
    const float* acting_table = (const float*)d_in[11];
    const float* nump_table   = (const float*)d_in[12];
    const float* W_scalar     = (const float*)d_in[13];
    const float* b_scalar     = (const float*)d_in[14];
    const float* W_bet        = (const float*)d_in[15];
    const float* b_bet        = (const float*)d_in[16];
    const float* W_action     = (const float*)d_in[17];
    const float* b_action     = (const float*)d_in[18];
    const float* W_combine    = (const float*)d_in[19];
    const float* b_combine    = (const float*)d_in[20];
    const float* ln_gamma     = (const float*)d_in[21];
    const float* ln_beta      = (const float*)d_in[22];

    // workspace layout (floats)
    float* ws   = (float*)d_ws;
    float* Tc   = ws;                    // 53*256
    float* Th   = Tc  + 53 * DM;         // 9*256
    float* Ta   = Th  + MP * DM;         // 9*256
    float* Tn   = Ta  + MP * DM;         // 10*256
    float* Ws2  = Tn  + (MP + 1) * DM;   // 2*256
    float* Wb2  = Ws2 + 2 * DM;          // 9*256
    float* Wa2  = Wb2 + MP * DM;         // 8*256
    float* btot = Wa2 + NA * DM;         // 256

    const dim3 blk32(32);
    // Tc = card_table @ W_combine[0:256]
    fused_table_gemm<<<dim3(4, 16),  blk32, 0, stream>>>(card_table,   W_combine + (size_t)   0 * DM, Tc,  53);
    // Th = hero_table @ W_combine[256:512]
    fused_table_gemm<<<dim3(1, 16),  blk32, 0, stream>>>(hero_table,   W_combine + (size_t) 256 * DM, Th,  MP);
    // Ta = acting_table @ W_combine[512:768]
    fused_table_gemm<<<dim3(1, 16),  blk32, 0, stream>>>(acting_table, W_combine + (size_t) 512 * DM, Ta,  MP);
    // Tn = nump_table @ W_combine[768:1024]
    fused_table_gemm<<<dim3(1, 16),  blk32, 0, stream>>>(nump_table,   W_combine + (size_t) 768 * DM, Tn,  MP + 1);
    // Ws2 = W_scalar @ W_combine[1024:1280]
    fused_table_gemm<<<dim3(1, 16),  blk32, 0, stream>>>(W_scalar,     W_combine + (size_t)1024 * DM, Ws2, 2);
    // Wb2 = W_bet @ W_combine[1280:1536]
    fused_table_gemm<<<dim3(1, 16),  blk32, 0, stream>>>(W_bet,        W_combine + (size_t)1280 * DM, Wb2, MP);
    // Wa2 = W_action @ W_combine[1536:1792]
    fused_table_gemm<<<dim3(1, 16),  blk32, 0, stream>>>(W_action,     W_combine + (size_t)1536 * DM, Wa2, NA);

    fuse_bias<<<1, 256, 0, stream>>>(b_combine, b_scalar, b_bet, b_action, W_combine, btot);

    float* out_emb  = (float*)d_out;
    float* out_mask = out_emb + (size_t)BB * EE * CC * DM;
    event_embed<<<BB * EE, CC * 32, 0, stream>>>(
        card_ids, hero_pos, acting_pos, num_players, seq_lengths,
        scalars, bets, action,
        Tc, Th, Ta, Tn, Ws2, Wb2, Wa2, btot,
        source_table, ln_gamma, ln_beta,
        out_emb, out_mask);
}